// MyModel_87522843559164
// MI455X (gfx1250) — compile-verified
//
#include <hip/hip_runtime.h>
#include <hip/hip_bf16.h>
#include <cstdint>

typedef __attribute__((ext_vector_type(2))) float v2f;
typedef __attribute__((ext_vector_type(8))) float v8f;

#define T_STEPS 24000
#define NGATE   400
#define UNITS_N 100
#define BN_EPS  1e-3f

__device__ __forceinline__ float fsig(float x)  { return 1.f / (1.f + __expf(-x)); }
__device__ __forceinline__ float ftanh(float x) { return 1.f - 2.f / (1.f + __expf(2.f * x)); }

// ---------------------------------------------------------------------------
// C[M,N] = A[M,K] @ B[K,N] + bias[N], fp32 via V_WMMA_F32_16X16X4_F32.
// One wave computes a 16x16 output tile; blockDim = (32,8) -> 8 M-tiles/block.
// A fragment (16x4 f32): lane = half*16 + m; v0/v1 = A[m][kb+2h], A[m][kb+2h+1]
// B fragment (4x16 f32): v0/v1 = B[kb+2h][n], B[kb+2h+1][n]
// C/D (16x16 f32): acc[v] = C[m0 + v + 8*half][n0 + (lane&15)]
// ---------------------------------------------------------------------------
__global__ void gemm_bias_wmma(const float* __restrict__ A,
                               const float* __restrict__ B,
                               const float* __restrict__ bias,
                               float* __restrict__ C,
                               int M, int N, int K) {
    const int lane = threadIdx.x;          // 0..31
    const int half = lane >> 4;            // 0/1
    const int l16  = lane & 15;
    const int n0 = blockIdx.x * 16;
    const int m0 = (blockIdx.y * 8 + threadIdx.y) * 16;
    if (m0 >= M) return;

    const float* __restrict__ Arow = A + (size_t)(m0 + l16) * K;
    const float* __restrict__ Bcol = B + (n0 + l16);

    v8f acc = {};
    const int K4 = K & ~3;
    int k = 0;
    for (; k < K4; k += 4) {
        const int ka = k + 2 * half;
        v2f a, b;
        a.x = Arow[ka];
        a.y = Arow[ka + 1];
        b.x = Bcol[(size_t)ka * N];
        b.y = Bcol[(size_t)(ka + 1) * N];
        acc = __builtin_amdgcn_wmma_f32_16x16x4_f32(false, a, false, b,
                                                    (short)0, acc, false, false);
    }
    if (k < K) {  // K % 4 != 0 tail (layer 1: K=598 -> tail of 2), zero-pad
        const int ka = k + 2 * half;
        v2f a, b;
        a.x = (ka     < K) ? Arow[ka]                  : 0.f;
        a.y = (ka + 1 < K) ? Arow[ka + 1]              : 0.f;
        b.x = (ka     < K) ? Bcol[(size_t)ka * N]      : 0.f;
        b.y = (ka + 1 < K) ? Bcol[(size_t)(ka + 1) * N] : 0.f;
        acc = __builtin_amdgcn_wmma_f32_16x16x4_f32(false, a, false, b,
                                                    (short)0, acc, false, false);
    }

    const float bv = bias[n0 + l16];
#pragma unroll
    for (int v = 0; v < 8; ++v) {
        const int m = m0 + v + 8 * half;
        C[(size_t)m * N + n0 + l16] = acc[v] + bv;
    }
}

// ---------------------------------------------------------------------------
// Sequential LSTM scan over T_STEPS for one layer, single workgroup (7 waves).
// Threads 0..199 each own 2 adjacent columns of U (100 x v2f, register-
// resident) and compute z[2m..2m+1] = xW[t] + h@U via packed f32 FMA.
// Threads 0..99 then run the gate nonlinearity, keep c in a register, publish
// raw h to LDS, and store BN(h) to global for the next layer's GEMM.
// ---------------------------------------------------------------------------
__global__ void __launch_bounds__(224, 1)
lstm_scan(const float* __restrict__ xW,   // [T_STEPS, NGATE]
          const float* __restrict__ Ug,   // [UNITS_N, NGATE]
          const float* __restrict__ gam,  // [UNITS_N]
          const float* __restrict__ bet,  // [UNITS_N]
          const float* __restrict__ mu,   // [UNITS_N]
          const float* __restrict__ var,  // [UNITS_N]
          float* __restrict__ hout)       // [T_STEPS, UNITS_N]  = BN(h)
{
    __shared__ __align__(16) float h_lds[UNITS_N];
    __shared__ __align__(16) float z_lds[NGATE];

    const int tid = threadIdx.x;
    const int m   = tid;            // owns gate columns 2m, 2m+1 when m < 200

    // Register-resident slice of U: column pair (2m, 2m+1), all 100 rows.
    v2f u[UNITS_N];
    if (m < 200) {
#pragma unroll
        for (int j = 0; j < UNITS_N; ++j)
            u[j] = *(const v2f*)(Ug + (size_t)j * NGATE + 2 * m);
    }

    if (tid < UNITS_N) h_lds[tid] = 0.f;
    float c = 0.f;
    float scale = 0.f, shift = 0.f;
    if (tid < UNITS_N) {
        scale = gam[tid] * rsqrtf(var[tid] + BN_EPS);
        shift = bet[tid] - mu[tid] * scale;
    }
    __syncthreads();

    for (int t = 0; t < T_STEPS; ++t) {
        if (m < 200) {
            const float* __restrict__ xr = xW + (size_t)t * NGATE + 2 * m;
            v2f zin = *(const v2f*)xr;                       // coalesced b64
            if (t + 16 < T_STEPS)
                __builtin_prefetch(xr + 16 * NGATE, 0, 0);   // global_prefetch_b8

            v2f a0 = {}, a1 = {}, a2 = {}, a3 = {};
#pragma unroll
            for (int j = 0; j < UNITS_N; j += 4) {
                const float4 hv = *(const float4*)(h_lds + j);  // LDS broadcast
                a0 += u[j]     * hv.x;                          // v_pk_fma_f32
                a1 += u[j + 1] * hv.y;
                a2 += u[j + 2] * hv.z;
                a3 += u[j + 3] * hv.w;
            }
            *(v2f*)(z_lds + 2 * m) = zin + (a0 + a1) + (a2 + a3);
        }
        __syncthreads();

        if (tid < UNITS_N) {
            const float zi = z_lds[tid];
            const float zf = z_lds[UNITS_N + tid];
            const float zg = z_lds[2 * UNITS_N + tid];
            const float zo = z_lds[3 * UNITS_N + tid];
            c = fsig(zf) * c + fsig(zi) * ftanh(zg);
            const float h = fsig(zo) * ftanh(c);
            h_lds[tid] = h;                                   // raw h recurrence
            hout[(size_t)t * UNITS_N + tid] = h * scale + shift;  // BN'd output
        }
        __syncthreads();
    }
}

// ---------------------------------------------------------------------------
// out[t] = softmax(h[t] @ Wd + bd), D_OUT = 3. One thread per timestep.
// ---------------------------------------------------------------------------
__global__ void dense_softmax(const float* __restrict__ h,   // [T, UNITS_N]
                              const float* __restrict__ Wd,  // [UNITS_N, 3]
                              const float* __restrict__ bd,  // [3]
                              float* __restrict__ out,       // [T, 3]
                              int T) {
    __shared__ float w[UNITS_N * 3];
    __shared__ float b[3];
    const int tid = threadIdx.x;
    for (int i = tid; i < UNITS_N * 3; i += blockDim.x) w[i] = Wd[i];
    if (tid < 3) b[tid] = bd[tid];
    __syncthreads();

    const int t = blockIdx.x * blockDim.x + tid;
    if (t >= T) return;

    float a0 = b[0], a1 = b[1], a2 = b[2];
    const float* __restrict__ hr = h + (size_t)t * UNITS_N;
#pragma unroll 4
    for (int j = 0; j < UNITS_N; ++j) {
        const float hv = hr[j];
        a0 = fmaf(hv, w[j * 3 + 0], a0);
        a1 = fmaf(hv, w[j * 3 + 1], a1);
        a2 = fmaf(hv, w[j * 3 + 2], a2);
    }
    const float mx = fmaxf(a0, fmaxf(a1, a2));
    const float e0 = __expf(a0 - mx), e1 = __expf(a1 - mx), e2 = __expf(a2 - mx);
    const float inv = 1.f / (e0 + e1 + e2);
    out[3 * t + 0] = e0 * inv;
    out[3 * t + 1] = e1 * inv;
    out[3 * t + 2] = e2 * inv;
}

// ---------------------------------------------------------------------------
extern "C" void kernel_launch(void* const* d_in, const int* in_sizes, int n_in,
                              void* d_out, int out_size, void* d_ws, size_t ws_size,
                              hipStream_t stream) {
    const float* x = (const float*)d_in[0];
    // per-layer params: W,U,b,g,be,m,v at 1+7*l ...
    const float* W[3]  = { (const float*)d_in[1],  (const float*)d_in[8],  (const float*)d_in[15] };
    const float* U[3]  = { (const float*)d_in[2],  (const float*)d_in[9],  (const float*)d_in[16] };
    const float* bb[3] = { (const float*)d_in[3],  (const float*)d_in[10], (const float*)d_in[17] };
    const float* g[3]  = { (const float*)d_in[4],  (const float*)d_in[11], (const float*)d_in[18] };
    const float* be[3] = { (const float*)d_in[5],  (const float*)d_in[12], (const float*)d_in[19] };
    const float* mm[3] = { (const float*)d_in[6],  (const float*)d_in[13], (const float*)d_in[20] };
    const float* vv[3] = { (const float*)d_in[7],  (const float*)d_in[14], (const float*)d_in[21] };
    const float* Wd = (const float*)d_in[22];
    const float* bd = (const float*)d_in[23];
    float* out = (float*)d_out;

    // Workspace: xW [24000,400] + two h ping-pong buffers [24000,100]
    float* xW = (float*)d_ws;
    float* hA = xW + (size_t)T_STEPS * NGATE;
    float* hB = hA + (size_t)T_STEPS * UNITS_N;

    const dim3 gblk(32, 8);
    const dim3 ggrid(NGATE / 16, (T_STEPS / 16 + 7) / 8);

    // Layer 1
    gemm_bias_wmma<<<ggrid, gblk, 0, stream>>>(x, W[0], bb[0], xW, T_STEPS, NGATE, 598);
    lstm_scan<<<1, 224, 0, stream>>>(xW, U[0], g[0], be[0], mm[0], vv[0], hA);
    // Layer 2
    gemm_bias_wmma<<<ggrid, gblk, 0, stream>>>(hA, W[1], bb[1], xW, T_STEPS, NGATE, UNITS_N);
    lstm_scan<<<1, 224, 0, stream>>>(xW, U[1], g[1], be[1], mm[1], vv[1], hB);
    // Layer 3
    gemm_bias_wmma<<<ggrid, gblk, 0, stream>>>(hB, W[2], bb[2], xW, T_STEPS, NGATE, UNITS_N);
    lstm_scan<<<1, 224, 0, stream>>>(xW, U[2], g[2], be[2], mm[2], vv[2], hA);
    // Head
    dense_softmax<<<(T_STEPS + 255) / 256, 256, 0, stream>>>(hA, Wd, bd, out, T_STEPS);
}